// HaarMSELoss_79044578115687
// MI455X (gfx1250) — compile-verified
//
#include <hip/hip_runtime.h>

// loss = (1/N_band) * sum_i (input_i - target_i)^2, N_band = numel/4
// (Haar 2x2 transform is orthonormal, so band-space SSE == pixel-space SSE.)

typedef __attribute__((ext_vector_type(4))) float f4;
typedef __attribute__((ext_vector_type(2))) float v2f;
typedef __attribute__((ext_vector_type(8))) float v8f;

#define NBLOCKS 2048
#define TPB 256
#define WAVES_PER_BLOCK (TPB / 32)

__device__ __forceinline__ float sq4(f4 x, f4 y, float acc) {
  f4 d = x - y;
  acc = fmaf(d.x, d.x, acc);
  acc = fmaf(d.y, d.y, acc);
  acc = fmaf(d.z, d.z, acc);
  acc = fmaf(d.w, d.w, acc);
  return acc;
}

__global__ __launch_bounds__(TPB) void haar_sse_partial_kernel(
    const float* __restrict__ in, const float* __restrict__ tg,
    float* __restrict__ partial, int nvec) {
  const f4* __restrict__ a = (const f4*)in;
  const f4* __restrict__ b = (const f4*)tg;

  int tid = blockIdx.x * TPB + threadIdx.x;
  int stride = gridDim.x * TPB;     // uniform (SGPR)
  int iters = nvec / stride;        // uniform trip count -> scalar loop control

  float acc0 = 0.0f, acc1 = 0.0f, acc2 = 0.0f, acc3 = 0.0f;
  int i = tid;
  int k = 0;
  // 4x unrolled: 8 independent global_load_b128 in flight per wave (4 KB)
  // before any consumer -> enough MLP to cover HBM latency at 23.3 TB/s.
  for (; k + 4 <= iters; k += 4) {
    f4 x0 = __builtin_nontemporal_load(&a[i]);
    f4 y0 = __builtin_nontemporal_load(&b[i]);
    f4 x1 = __builtin_nontemporal_load(&a[i + stride]);
    f4 y1 = __builtin_nontemporal_load(&b[i + stride]);
    f4 x2 = __builtin_nontemporal_load(&a[i + 2 * stride]);
    f4 y2 = __builtin_nontemporal_load(&b[i + 2 * stride]);
    f4 x3 = __builtin_nontemporal_load(&a[i + 3 * stride]);
    f4 y3 = __builtin_nontemporal_load(&b[i + 3 * stride]);
    acc0 = sq4(x0, y0, acc0);
    acc1 = sq4(x1, y1, acc1);
    acc2 = sq4(x2, y2, acc2);
    acc3 = sq4(x3, y3, acc3);
    i += 4 * stride;
  }
  for (; k < iters; ++k) {  // leftover full sweeps (uniform)
    acc0 = sq4(__builtin_nontemporal_load(&a[i]),
               __builtin_nontemporal_load(&b[i]), acc0);
    i += stride;
  }
  if (i < nvec) {  // ragged tail (empty for this problem size)
    acc1 = sq4(__builtin_nontemporal_load(&a[i]),
               __builtin_nontemporal_load(&b[i]), acc1);
  }
  float acc = (acc0 + acc1) + (acc2 + acc3);

  // ---- Wave32 reduction via V_WMMA_F32_16X16X4_F32 ----
  // A (16x4 f32, 2 VGPRs/lane): lane partial in K-slot 0, zero in slot 1.
  // B = all ones. D[m][n] = rowsum_m:
  //   lanes 0..15:  C[j] = rowsum(j);  lanes 16..31: C[j] = rowsum(j+8)
  // sum(C[0..7]) + shfl_down(.,16) => 32-lane sum in lanes 0..15.
  // EXEC is all-1s here: every thread of the block reaches this point.
  v2f A; A[0] = acc;  A[1] = 0.0f;
  v2f B; B[0] = 1.0f; B[1] = 1.0f;
  v8f C = {};
  C = __builtin_amdgcn_wmma_f32_16x16x4_f32(
      /*neg_a=*/false, A, /*neg_b=*/false, B,
      /*c_mod=*/(short)0, C, /*reuse_a=*/false, /*reuse_b=*/false);

  float s = ((C[0] + C[1]) + (C[2] + C[3])) + ((C[4] + C[5]) + (C[6] + C[7]));
  s += __shfl_down(s, 16, 32);

  __shared__ float wsum[WAVES_PER_BLOCK];
  int lane = threadIdx.x & 31;
  int wave = threadIdx.x >> 5;
  if (lane == 0) wsum[wave] = s;
  __syncthreads();
  if (threadIdx.x == 0) {
    float t = 0.0f;
#pragma unroll
    for (int w = 0; w < WAVES_PER_BLOCK; ++w) t += wsum[w];
    partial[blockIdx.x] = t;  // deterministic per-block partial
  }
}

__global__ __launch_bounds__(256) void haar_sse_finalize_kernel(
    const float* __restrict__ partial, float* __restrict__ out,
    int n, float scale) {
  __shared__ float sm[256];
  float t = 0.0f;
  for (int i = threadIdx.x; i < n; i += 256) t += partial[i];
  sm[threadIdx.x] = t;
  __syncthreads();
#pragma unroll
  for (int off = 128; off > 0; off >>= 1) {
    if (threadIdx.x < off) sm[threadIdx.x] += sm[threadIdx.x + off];
    __syncthreads();
  }
  if (threadIdx.x == 0) out[0] = sm[0] * scale;
}

extern "C" void kernel_launch(void* const* d_in, const int* in_sizes, int n_in,
                              void* d_out, int out_size, void* d_ws, size_t ws_size,
                              hipStream_t stream) {
  const float* in = (const float*)d_in[0];
  const float* tg = (const float*)d_in[1];
  float* out = (float*)d_out;
  float* ws = (float*)d_ws;  // NBLOCKS floats of scratch (8 KiB)

  int numel = in_sizes[0];            // 4*32*512*512 = 33,554,432
  int nvec = numel >> 2;              // float4 count (numel divisible by 4)
  float scale = 4.0f / (float)numel;  // 1/N_band, N_band = numel/4

  haar_sse_partial_kernel<<<NBLOCKS, TPB, 0, stream>>>(in, tg, ws, nvec);
  haar_sse_finalize_kernel<<<1, 256, 0, stream>>>(ws, out, NBLOCKS, scale);
}